// FastSeqProp_7799660610237
// MI455X (gfx1250) — compile-verified
//
#include <hip/hip_runtime.h>
#include <stdint.h>

// Problem shape (from reference): N=8192, C=4, L=200, padded length 600.
#define CCH 4
#define LL  200
#define LPAD 600
#define EPSV 1e-5f

// Counter-based PRNG (deterministic, keyed): PCG output function.
__device__ __forceinline__ unsigned pcg_hash(unsigned v) {
  unsigned s = v * 747796405u + 2891336453u;
  unsigned w = ((s >> ((s >> 28u) + 4u)) ^ s) * 277803737u;
  return (w >> 22u) ^ w;
}

__global__ __launch_bounds__(256) void fastseqprop_kernel(
    const float* __restrict__ seq,    // (N,4,200)
    const float* __restrict__ scale,  // (N,4,1)
    const float* __restrict__ shift,  // (N,4,1)
    const float* __restrict__ up,     // (N,4,200)
    const float* __restrict__ dn,     // (N,4,200)
    float* __restrict__ out0,         // (N,4,600) softmaxed (padded)
    float* __restrict__ out1) {       // (N,4,600) sampled   (padded)
  __shared__ float tile[CCH * LL];    // 800 floats: the whole (4x200) slab
  __shared__ float wsum[8][CCH];
  __shared__ float wsq[8][CCH];
  __shared__ float sA[CCH];           // rstd * scale
  __shared__ float sB[CCH];           // shift - mean * rstd * scale

  const int tid = threadIdx.x;
  const int n = blockIdx.x;
  const size_t nCL = (size_t)n * (CCH * LL);
  const size_t nOut = (size_t)n * (CCH * LPAD);

  // ---- Stage the sequence tile into LDS with gfx1250 async copy engine.
  // 200 lanes x B128 = 800 floats; 16B aligned (base n*3200B, step 16B).
  if (tid < LL) {
    unsigned loff = (unsigned)(uintptr_t)(&tile[0]) + (unsigned)tid * 16u;
    uint64_t gaddr = (uint64_t)(uintptr_t)(seq + nCL + (size_t)tid * 4);
    asm volatile("global_load_async_to_lds_b128 %0, %1, off"
                 :: "v"(loff), "v"(gaddr) : "memory");
  }

  // ---- Overlap: stream the pad regions to BOTH outputs while the async
  // copy is in flight (pure HBM streaming, independent of LDS contents).
  if (tid < 200) {
    const float4* up4 = (const float4*)(up + nCL);
    const float4* dn4 = (const float4*)(dn + nCL);
    int c = tid / 50;           // channel
    int r = tid - c * 50;       // float4 index within the 200-float segment
    float4 u4 = up4[tid];
    float4 d4 = dn4[tid];
    float* r0 = out0 + nOut + (size_t)c * LPAD;
    float* r1 = out1 + nOut + (size_t)c * LPAD;
    ((float4*)r0)[r]         = u4;   // cols [0,200)
    ((float4*)(r0 + 400))[r] = d4;   // cols [400,600)
    ((float4*)r1)[r]         = u4;
    ((float4*)(r1 + 400))[r] = d4;
  }

  // ---- Join the async copy, then make LDS visible work-group wide.
  asm volatile("s_wait_asynccnt 0x0" ::: "memory");
  __syncthreads();

  // ---- Per-position values (thread tid owns column l=tid).
  float x[CCH];
#pragma unroll
  for (int c = 0; c < CCH; ++c) x[c] = (tid < LL) ? tile[c * LL + tid] : 0.f;

  // ---- Per-channel sum / sumsq over L: wave32 shuffle tree + LDS partials.
  float s[CCH], q[CCH];
#pragma unroll
  for (int c = 0; c < CCH; ++c) { s[c] = x[c]; q[c] = x[c] * x[c]; }
#pragma unroll
  for (int m = 16; m >= 1; m >>= 1) {
#pragma unroll
    for (int c = 0; c < CCH; ++c) {
      s[c] += __shfl_xor(s[c], m);
      q[c] += __shfl_xor(q[c], m);
    }
  }
  const int wave = tid >> 5;
  if ((tid & 31) == 0) {
#pragma unroll
    for (int c = 0; c < CCH; ++c) { wsum[wave][c] = s[c]; wsq[wave][c] = q[c]; }
  }
  __syncthreads();

  if (tid < CCH) {
    float ts = 0.f, tq = 0.f;
#pragma unroll
    for (int w = 0; w < 8; ++w) { ts += wsum[w][tid]; tq += wsq[w][tid]; }
    const float invL = 1.f / (float)LL;
    float mean = ts * invL;
    float var = fmaf(tq, invL, -mean * mean);   // biased variance
    float rstd = rsqrtf(var + EPSV);
    float g = scale[(size_t)n * CCH + tid];
    float b = shift[(size_t)n * CCH + tid];
    sA[tid] = rstd * g;
    sB[tid] = fmaf(-mean * rstd, g, b);
  }
  __syncthreads();

  // ---- Normalize+affine, softmax over 4 channels, categorical sample, STE.
  if (tid < LL) {
    float sc[CCH];
#pragma unroll
    for (int c = 0; c < CCH; ++c) sc[c] = fmaf(x[c], sA[c], sB[c]);

    float mx = fmaxf(fmaxf(sc[0], sc[1]), fmaxf(sc[2], sc[3]));
    float e[CCH], sum = 0.f;
#pragma unroll
    for (int c = 0; c < CCH; ++c) { e[c] = __expf(sc[c] - mx); sum += e[c]; }
    float inv = 1.f / sum;
    float p[CCH];
#pragma unroll
    for (int c = 0; c < CCH; ++c) p[c] = e[c] * inv;

    // Deterministic categorical draw keyed on (seed=42, n*L + l).
    unsigned idx = (unsigned)(n * LL + tid);
    unsigned h = pcg_hash(idx + 42u * 0x9E3779B9u);
    float u = (float)(h >> 8) * (1.f / 16777216.f);
    float c0 = p[0];
    float c1 = c0 + p[1];
    float c2 = c1 + p[2];
    int k = (u >= c0) + (u >= c1) + (u >= c2);

#pragma unroll
    for (int c = 0; c < CCH; ++c) {
      float pv = p[c];
      float oh = (c == k) ? 1.f : 0.f;
      size_t o = nOut + (size_t)c * LPAD + LL + tid;   // middle cols [200,400)
      out0[o] = pv;
      out1[o] = (oh - pv) + pv;   // straight-through estimator arithmetic
    }
  }
}

extern "C" void kernel_launch(void* const* d_in, const int* in_sizes, int n_in,
                              void* d_out, int out_size, void* d_ws, size_t ws_size,
                              hipStream_t stream) {
  const float* seq   = (const float*)d_in[0];
  const float* scale = (const float*)d_in[1];
  const float* shift = (const float*)d_in[2];
  const float* up    = (const float*)d_in[3];
  const float* dn    = (const float*)d_in[4];
  const int N = in_sizes[0] / (CCH * LL);   // 8192
  float* out0 = (float*)d_out;
  float* out1 = out0 + (size_t)N * CCH * LPAD;
  fastseqprop_kernel<<<dim3(N), dim3(256), 0, stream>>>(
      seq, scale, shift, up, dn, out0, out1);
}